// CTCLayer_6803228197119
// MI455X (gfx1250) — compile-verified
//
#include <hip/hip_runtime.h>
#include <stdint.h>

// Problem constants (match reference setup_inputs)
#define B_    128
#define T_    512
#define C_    1024
#define L_    64
#define BLANK_ (C_ - 1)
#define S_    (2 * L_ + 1)   // 129 extended positions
#define SPAD  132            // padded alpha length
#define PF    8              // async prefetch depth (power of 2)
#define NEGV  (-1.0e9f)
#define EPS_  (1e-7f)
#define NTHREADS 160         // 5 wave32 waves, covers S_=129

typedef __attribute__((address_space(1))) int as1_int;  // global
typedef __attribute__((address_space(3))) int as3_int;  // LDS

// Per-lane async gather: copies 4 bytes from the global address gptr to the LDS address lptr via the CDNA5 async engine
// (GLOBAL_LOAD_ASYNC_TO_LDS_B32, tracked with ASYNCcnt).
__device__ __forceinline__ void async_gather_b32(const float* gptr, float* lptr) {
#if __has_builtin(__builtin_amdgcn_global_load_async_to_lds_b32)
    __builtin_amdgcn_global_load_async_to_lds_b32(
        (as1_int*)(uintptr_t)gptr,
        (as3_int*)(uint32_t)(uintptr_t)lptr,  // low 32 bits of flat LDS addr = LDS offset
        0, 0);
#else
    asm volatile("global_load_async_to_lds_b32 %0, %1, off"
                 :: "v"((uint32_t)(uintptr_t)lptr), "v"((uint64_t)(uintptr_t)gptr)
                 : "memory");
#endif
}

// Allow up to PF-1 newer async ops outstanding -> oldest needed slot complete.
__device__ __forceinline__ void wait_asynccnt_le(void) {
#if __has_builtin(__builtin_amdgcn_s_wait_asynccnt)
    __builtin_amdgcn_s_wait_asynccnt(PF - 1);
#else
    asm volatile("s_wait_asynccnt 7" ::: "memory");
#endif
}

__global__ __launch_bounds__(NTHREADS)
void ctc_loss_kernel(const int* __restrict__ y_true,
                     const float* __restrict__ y_pred,
                     const int* __restrict__ input_length,
                     const int* __restrict__ label_length,
                     float* __restrict__ out)
{
    __shared__ float A[2][SPAD];      // double-buffered alpha
    __shared__ float E[PF][SPAD];     // async-gathered emissions, PF slots
    __shared__ int   ylab[L_];        // this batch's label row

    const int b = blockIdx.x;
    const int s = threadIdx.x;

    const int in_len  = input_length[b];
    const int lab_len = label_length[b];

    if (s < L_)   ylab[s] = y_true[(size_t)b * L_ + s];
    if (s < SPAD) A[0][s] = (s == 0) ? 0.0f : NEGV;
    __syncthreads();

    // Per-lane static state: extended label class and skip-transition flag.
    int  lab = BLANK_;
    bool skip_ok = false;
    if (s < S_) {
        if (s & 1) {
            lab = ylab[(s - 1) >> 1];                 // labels are < BLANK by construction
            if (s >= 3) skip_ok = (lab != ylab[(s - 3) >> 1]);
        }
    }
    const bool valid = (s < 2 * lab_len + 1);
    const float* gbase = y_pred + (size_t)b * T_ * C_ + lab;  // lane's column; stride C_ per t

    // Prologue: fill the PF-deep async pipeline.
    if (s < S_) {
        #pragma unroll
        for (int tp = 0; tp < PF; ++tp)
            async_gather_b32(gbase + (size_t)tp * C_, &E[tp][s]);
    }

    // Prime the pipelined emission log for t = 0.
    wait_asynccnt_le();                      // slot 0 complete (PF-1 newer outstanding)
    float e = 0.0f;
    if (s < S_) e = __logf(E[0][s] + EPS_);

    int p = 0;
    for (int t = 0; t < T_; ++t) {
        float e_next = 0.0f;
        if (s < S_) {
            const float a0 = A[p][s];
            const float a1 = (s >= 1) ? A[p][s - 1] : NEGV;
            const float a2 = skip_ok  ? A[p][s - 2] : NEGV;
            const float m  = fmaxf(a0, fmaxf(a1, a2));
            const float lse = m + __logf(__expf(a0 - m) + __expf(a1 - m) + __expf(a2 - m));
            float an = e + lse;
            an = valid ? an : NEGV;
            an = (t < in_len) ? an : a0;     // freeze past per-example input length
            A[p ^ 1][s] = an;
            if (t + PF < T_)                 // refill the slot consumed at step t
                async_gather_b32(gbase + (size_t)(t + PF) * C_, &E[t & (PF - 1)][s]);
        }
        wait_asynccnt_le();                  // slot t+1 complete
        if (s < S_)                          // next emission log in the barrier shadow
            e_next = __logf(E[(t + 1) & (PF - 1)][s] + EPS_);
        __syncthreads();                     // one barrier per timestep
        e = e_next;
        p ^= 1;
    }

    if (s == 0) {
        const int   eb = 2 * lab_len;
        const float a1 = A[p][eb];
        const float a2 = (eb >= 1) ? A[p][eb - 1] : NEGV;
        const float m  = fmaxf(a1, a2);
        out[b] = -(m + __logf(__expf(a1 - m) + __expf(a2 - m)));
    }
}

extern "C" void kernel_launch(void* const* d_in, const int* in_sizes, int n_in,
                              void* d_out, int out_size, void* d_ws, size_t ws_size,
                              hipStream_t stream)
{
    (void)in_sizes; (void)n_in; (void)out_size; (void)d_ws; (void)ws_size;
    const int*   y_true  = (const int*)  d_in[0];
    const float* y_pred  = (const float*)d_in[1];
    const int*   in_len  = (const int*)  d_in[2];
    const int*   lab_len = (const int*)  d_in[3];
    ctc_loss_kernel<<<B_, NTHREADS, 0, stream>>>(y_true, y_pred, in_len, lab_len,
                                                 (float*)d_out);
}